// EvolveGCN_47596827574587
// MI455X (gfx1250) — compile-verified
//
#include <hip/hip_runtime.h>
#include <hip/hip_bf16.h>

#define D_EMB 64
#define D2    4096       // 64*64
#define GATE  12288      // 3*D2

typedef __attribute__((ext_vector_type(2))) float v2f;
typedef __attribute__((ext_vector_type(8))) float v8f;

__device__ __forceinline__ void atomicAddF32(float* p, float v) {
    __hip_atomic_fetch_add(p, v, __ATOMIC_RELAXED, __HIP_MEMORY_SCOPE_AGENT);
}

// ---------------------------------------------------------------------------
// Batched GRU gate matvec: for each row r of w_ih/w_hh (streamed once),
// compute dot with both GRU inputs (x1,x2) and both hidden states (h1,h2).
// One block per row, 256 threads, LDS tree reduction of 4 partials.
// 400 MB of weight traffic total -> ~17 us at 23.3 TB/s; pure bandwidth.
// ---------------------------------------------------------------------------
__global__ void gru_matvec_kernel(const float* __restrict__ w_ih,
                                  const float* __restrict__ w_hh,
                                  const float* __restrict__ x1,
                                  const float* __restrict__ x2,
                                  const float* __restrict__ h1,
                                  const float* __restrict__ h2,
                                  const float* __restrict__ b_ih,
                                  const float* __restrict__ b_hh,
                                  float* __restrict__ gi1, float* __restrict__ gi2,
                                  float* __restrict__ gh1, float* __restrict__ gh2) {
    __shared__ float red[4][256];
    const int r = blockIdx.x;
    const int t = threadIdx.x;
    const float* wi = w_ih + (size_t)r * D2;
    const float* wh = w_hh + (size_t)r * D2;
    float ai1 = 0.f, ai2 = 0.f, ah1 = 0.f, ah2 = 0.f;
    for (int j = t; j < D2; j += 256) {
        const float a = wi[j];
        const float b = wh[j];
        ai1 += a * x1[j];  ai2 += a * x2[j];
        ah1 += b * h1[j];  ah2 += b * h2[j];
    }
    red[0][t] = ai1; red[1][t] = ai2; red[2][t] = ah1; red[3][t] = ah2;
    __syncthreads();
    for (int s = 128; s > 0; s >>= 1) {
        if (t < s) {
            red[0][t] += red[0][t + s];
            red[1][t] += red[1][t + s];
            red[2][t] += red[2][t + s];
            red[3][t] += red[3][t + s];
        }
        __syncthreads();
    }
    if (t == 0) {
        gi1[r] = red[0][0] + b_ih[r];
        gi2[r] = red[1][0] + b_ih[r];
        gh1[r] = red[2][0] + b_hh[r];
        gh2[r] = red[3][0] + b_hh[r];
    }
}

// ---------------------------------------------------------------------------
// GRU gate combine: r/z/n gates -> evolved weight matrices w1, w2 (flat 4096).
// ---------------------------------------------------------------------------
__global__ void gru_combine_kernel(const float* __restrict__ gi1,
                                   const float* __restrict__ gi2,
                                   const float* __restrict__ gh1,
                                   const float* __restrict__ gh2,
                                   const float* __restrict__ h1,
                                   const float* __restrict__ h2,
                                   float* __restrict__ w1,
                                   float* __restrict__ w2) {
    const int idx = blockIdx.x * blockDim.x + threadIdx.x;
    if (idx >= 2 * D2) return;
    const int g = idx >> 12;
    const int j = idx & (D2 - 1);
    const float* gi = g ? gi2 : gi1;
    const float* gh = g ? gh2 : gh1;
    const float* h  = g ? h2  : h1;
    float*       w  = g ? w2  : w1;
    const float rg = 1.f / (1.f + __expf(-(gi[j] + gh[j])));
    const float zg = 1.f / (1.f + __expf(-(gi[D2 + j] + gh[D2 + j])));
    const float ng = tanhf(gi[2 * D2 + j] + rg * gh[2 * D2 + j]);
    w[j] = (1.f - zg) * ng + zg * h[j];
}

// ---------------------------------------------------------------------------
// Helpers: zero buffer, degree count, rsqrt norm, edge norm.
// ---------------------------------------------------------------------------
__global__ void zero_kernel(float* __restrict__ p, int n) {
    const int i = blockIdx.x * blockDim.x + threadIdx.x;
    if (i < n) p[i] = 0.f;
}

__global__ void degree_kernel(const int* __restrict__ dst, float* __restrict__ deg, int E) {
    const int e = blockIdx.x * blockDim.x + threadIdx.x;
    if (e < E) atomicAddF32(&deg[dst[e]], 1.0f);
}

__global__ void norm_kernel(float* __restrict__ deg, int N) {
    const int i = blockIdx.x * blockDim.x + threadIdx.x;
    if (i < N) {
        const float d = deg[i];
        deg[i] = rsqrtf(d < 1.f ? 1.f : d);   // clip(deg,1)^-0.5, in place
    }
}

__global__ void edgenorm_kernel(const int* __restrict__ src, const int* __restrict__ dst,
                                const float* __restrict__ norm, float* __restrict__ en, int E) {
    const int e = blockIdx.x * blockDim.x + threadIdx.x;
    if (e < E) en[e] = norm[src[e]] * norm[dst[e]];
}

// ---------------------------------------------------------------------------
// Dense transform C[M x 64] = act(X)[M x 64] @ W[64 x 64] with
// V_WMMA_F32_16X16X4_F32 (fp32 matrix core, full precision).
// Block = 128 threads = 4 wave32s; block owns one 16-row M tile, each wave one
// 16-col N tile; 16 K-steps of 4.
//   A frag (16x4, ISA 7.12.2): lane<16 -> (M=lane, K=0..1), lane>=16 ->
//   (M=lane-16, K=2..3); loaded as one aligned float2 (global_load_b64).
//   B frag (4x16): v0/v1 = K rows {2*half, 2*half+1} striped over lane halves;
//   all 16 fragments are loop-invariant -> hoisted into registers (32 VGPRs).
//   C/D: row = half*8 + v.
// RELU is a template param: layer-2 instantiation gets a bare v_max_num_f32,
// no runtime cndmask in the hot loop.
// ---------------------------------------------------------------------------
template <bool RELU>
__global__ void gcn_gemm_wmma_kernel(const float* __restrict__ X,
                                     const float* __restrict__ W,
                                     float* __restrict__ C,
                                     int M) {
    const int wave = threadIdx.x >> 5;
    const int lane = threadIdx.x & 31;
    const int m0   = blockIdx.x * 16;
    const int n0   = wave * 16;
    const int half = lane >> 4;         // 0: K pair {0,1}, 1: K pair {2,3}
    const int l    = lane & 15;

    // Hoist all 16 B fragments (the wave's 64x16 W tile) out of the K loop.
    v2f b[16];
    #pragma unroll
    for (int kk = 0; kk < 16; ++kk) {
        const float* wp = W + (kk * 4 + 2 * half) * D_EMB + n0 + l;
        b[kk].x = wp[0];
        b[kk].y = wp[D_EMB];
    }

    int arow = m0 + l;
    if (arow > M - 1) arow = M - 1;     // clamp (M=100000 divisible; safety)
    // Row is 256B-aligned; pair offset (kk*2 + half) is 8B-aligned -> b64 loads.
    const v2f* xp = (const v2f*)(X + (size_t)arow * D_EMB) + half;

    v8f acc = {};
    #pragma unroll
    for (int kk = 0; kk < 16; ++kk) {
        v2f a = xp[kk * 2];
        if (RELU) {
            a.x = fmaxf(a.x, 0.f);
            a.y = fmaxf(a.y, 0.f);
        }
        acc = __builtin_amdgcn_wmma_f32_16x16x4_f32(
            /*neg_a=*/false, a, /*neg_b=*/false, b[kk],
            /*c_mod=*/(short)0, acc, /*reuse_a=*/false, /*reuse_b=*/false);
    }

    #pragma unroll
    for (int v = 0; v < 8; ++v) {
        const int row = m0 + half * 8 + v;
        if (row < M) C[(size_t)row * D_EMB + n0 + l] = acc[v];
    }
}

// ---------------------------------------------------------------------------
// Edge scatter: out[dst[e], :] += h[src[e], :] * en[e].
// One wave32 per edge; lane handles features {lane, lane+32}; native
// global_atomic_add_f32 (relaxed/agent). ~820 MB traffic per layer -> the
// dominant, bandwidth-bound stage.
// ---------------------------------------------------------------------------
__global__ void scatter_kernel(const float* __restrict__ h,
                               const float* __restrict__ en,
                               const int* __restrict__ src,
                               const int* __restrict__ dst,
                               float* __restrict__ out, int E) {
    const int wave = (blockIdx.x * blockDim.x + threadIdx.x) >> 5;
    const int lane = threadIdx.x & 31;
    if (wave >= E) return;
    const int s = src[wave];
    const int d = dst[wave];
    const float w = en[wave];
    const float v0 = h[(size_t)s * D_EMB + lane]      * w;
    const float v1 = h[(size_t)s * D_EMB + lane + 32] * w;
    atomicAddF32(&out[(size_t)d * D_EMB + lane],      v0);
    atomicAddF32(&out[(size_t)d * D_EMB + lane + 32], v1);
}

// ---------------------------------------------------------------------------
extern "C" void kernel_launch(void* const* d_in, const int* in_sizes, int n_in,
                              void* d_out, int out_size, void* d_ws, size_t ws_size,
                              hipStream_t stream) {
    const float* node_emb = (const float*)d_in[0];
    const float* gc1_w    = (const float*)d_in[1];
    const float* gc2_w    = (const float*)d_in[2];
    const float* gc1_h    = (const float*)d_in[3];
    const float* gc2_h    = (const float*)d_in[4];
    const float* w_ih     = (const float*)d_in[5];
    const float* w_hh     = (const float*)d_in[6];
    const float* b_ih     = (const float*)d_in[7];
    const float* b_hh     = (const float*)d_in[8];
    const int*   src      = (const int*)d_in[9];
    const int*   dst      = (const int*)d_in[10];

    const int N  = in_sizes[0] / D_EMB;   // 100000
    const int E  = in_sizes[9];           // 1600000
    const int NF = N * D_EMB;             // 6.4M floats

    // Workspace layout (floats), ~58 MB total
    float* ws  = (float*)d_ws;
    float* gi1 = ws;                 // GATE
    float* gi2 = gi1 + GATE;
    float* gh1 = gi2 + GATE;
    float* gh2 = gh1 + GATE;
    float* w1  = gh2 + GATE;         // D2
    float* w2  = w1 + D2;
    float* deg = w2 + D2;            // N (reused as norm)
    float* en  = deg + N;            // E
    float* H   = en + E;             // N*64 (reused for layer-2 transform)
    float* A1  = H + (size_t)NF;     // N*64
    float* out = (float*)d_out;

    const int mtiles = (N + 15) / 16;

    // --- GRU weight evolution ---
    gru_matvec_kernel<<<GATE, 256, 0, stream>>>(w_ih, w_hh, gc1_h, gc2_h,
                                                gc1_w, gc2_w, b_ih, b_hh,
                                                gi1, gi2, gh1, gh2);
    gru_combine_kernel<<<(2 * D2 + 255) / 256, 256, 0, stream>>>(
        gi1, gi2, gh1, gh2, gc1_w, gc2_w, w1, w2);

    // --- degree normalization ---
    zero_kernel<<<(N + 255) / 256, 256, 0, stream>>>(deg, N);
    degree_kernel<<<(E + 255) / 256, 256, 0, stream>>>(dst, deg, E);
    norm_kernel<<<(N + 255) / 256, 256, 0, stream>>>(deg, N);
    edgenorm_kernel<<<(E + 255) / 256, 256, 0, stream>>>(src, dst, deg, en, E);

    // --- layer 1: H = node_emb @ w1 ; A1 = scatter(H) ---
    gcn_gemm_wmma_kernel<false><<<mtiles, 128, 0, stream>>>(node_emb, w1, H, N);
    zero_kernel<<<(NF + 255) / 256, 256, 0, stream>>>(A1, NF);
    scatter_kernel<<<(E + 7) / 8, 256, 0, stream>>>(H, en, src, dst, A1, E);

    // --- layer 2: H = relu(A1) @ w2 ; out = scatter(H) ---
    gcn_gemm_wmma_kernel<true><<<mtiles, 128, 0, stream>>>(A1, w2, H, N);
    zero_kernel<<<(NF + 255) / 256, 256, 0, stream>>>(out, NF);
    scatter_kernel<<<(E + 7) / 8, 256, 0, stream>>>(H, en, src, dst, out, E);
}